// Net_31404800868533
// MI455X (gfx1250) — compile-verified
//
#include <hip/hip_runtime.h>
#include <hip/hip_bf16.h>
#include <math.h>

// ---------------------------------------------------------------------------
// GAT pipeline for MI455X (gfx1250, wave32).
//   - Dense projections: V_WMMA_F32_16X16X4_F32, templated K/NC so the inner
//     loop is branch-free, A-frag loaded as b64, 4 accumulators per wave.
//   - Edge softmax/scatter: wave-per-edge, float4 gathers, f32 global atomics.
// ---------------------------------------------------------------------------

#define NNODES 20000
#define NEDGES 320000
#define FIN    33
#define HF     64      // per-head feature dim (D) -- constant across all layers
#define NHEADS 4
#define HDIM   256     // NHEADS * HF
#define BN_EPS 1e-5f

typedef __attribute__((ext_vector_type(2))) float v2f;
typedef __attribute__((ext_vector_type(8))) float v8f;

__device__ __forceinline__ v8f wmma4(v2f a, v2f b, v8f c) {
  // (neg_a, A, neg_b, B, c_mod, C, reuse_a, reuse_b)
  return __builtin_amdgcn_wmma_f32_16x16x4_f32(false, a, false, b,
                                               (short)0, c, false, false);
}

// ------------------------- WMMA f32 GEMM -----------------------------------
// C[M,NC] = A[M,K] @ W[K,NC] (+bias).  One wave computes a 16x64 strip:
// A-frag loaded once per K-step, reused by 4 WMMAs (4 accumulators).
// grid = (M/16, NC/64).  K, NC compile-time -> no predication in main loop.
template<int K, int NC>
__global__ __launch_bounds__(32)
void gemm_wmma(const float* __restrict__ A, const float* __restrict__ W,
               const float* __restrict__ bias, float* __restrict__ C)
{
  const int lane = threadIdx.x & 31;
  const int half = lane >> 4;
  const int lo   = lane & 15;
  const int row0 = blockIdx.x * 16;
  const int col0 = blockIdx.y * 64;

  constexpr int K4 = K & ~3;
  const float* __restrict__ arow = A + (size_t)(row0 + lo) * K;

  v8f acc0 = {}, acc1 = {}, acc2 = {}, acc3 = {};

#pragma unroll 4
  for (int k0 = 0; k0 < K4; k0 += 4) {
    const int ka = k0 + 2 * half;
    v2f av;
    if constexpr ((K & 1) == 0) {
      av = *(const v2f*)(arow + ka);          // 8B-aligned b64 load
    } else {
      av.x = arow[ka];
      av.y = arow[ka + 1];
    }
    const float* __restrict__ w0 = W + (size_t)ka * NC + col0 + lo;
    const float* __restrict__ w1 = w0 + NC;
    v2f b0, b1, b2, b3;
    b0.x = w0[0];  b0.y = w1[0];
    b1.x = w0[16]; b1.y = w1[16];
    b2.x = w0[32]; b2.y = w1[32];
    b3.x = w0[48]; b3.y = w1[48];
    acc0 = wmma4(av, b0, acc0);
    acc1 = wmma4(av, b1, acc1);
    acc2 = wmma4(av, b2, acc2);
    acc3 = wmma4(av, b3, acc3);
  }

  if constexpr ((K & 3) != 0) {
    // tail K-step: clamp indices in-bounds, mask A with cndmask (no branches).
    const int ka = K4 + 2 * half;
    const bool p0 = (ka     < K);
    const bool p1 = (ka + 1 < K);
    const int i0 = p0 ? ka     : (K - 1);
    const int i1 = p1 ? ka + 1 : (K - 1);
    const float a0 = arow[i0];
    const float a1 = arow[i1];
    v2f av;
    av.x = p0 ? a0 : 0.0f;
    av.y = p1 ? a1 : 0.0f;
    const float* __restrict__ w0 = W + (size_t)i0 * NC + col0 + lo;
    const float* __restrict__ w1 = W + (size_t)i1 * NC + col0 + lo;
    v2f b0, b1, b2, b3;
    b0.x = w0[0];  b0.y = w1[0];
    b1.x = w0[16]; b1.y = w1[16];
    b2.x = w0[32]; b2.y = w1[32];
    b3.x = w0[48]; b3.y = w1[48];
    acc0 = wmma4(av, b0, acc0);
    acc1 = wmma4(av, b1, acc1);
    acc2 = wmma4(av, b2, acc2);
    acc3 = wmma4(av, b3, acc3);
  }

  float bc0 = 0.0f, bc1 = 0.0f, bc2 = 0.0f, bc3 = 0.0f;
  if (bias) {
    bc0 = bias[col0 + lo];
    bc1 = bias[col0 + lo + 16];
    bc2 = bias[col0 + lo + 32];
    bc3 = bias[col0 + lo + 48];
  }
#pragma unroll
  for (int v = 0; v < 8; ++v) {
    const int row = row0 + v + 8 * half;
    float* __restrict__ crow = C + (size_t)row * NC + col0 + lo;
    crow[0]  = acc0[v] + bc0;
    crow[16] = acc1[v] + bc1;
    crow[32] = acc2[v] + bc2;
    crow[48] = acc3[v] + bc3;
  }
}

// ------------------------- BatchNorm (training stats) ----------------------
__global__ __launch_bounds__(64)
void bn_stats_kernel(const float* __restrict__ x, float* __restrict__ stats, int Nrows)
{
  const int col = threadIdx.x;                    // blockDim.x == HF
  float s = 0.0f, s2 = 0.0f;
  for (int r = blockIdx.x; r < Nrows; r += gridDim.x) {
    const float v = x[(size_t)r * HF + col];
    s += v; s2 += v * v;
  }
  atomicAdd(&stats[col], s);
  atomicAdd(&stats[HF + col], s2);
}

__global__ __launch_bounds__(256)
void bn_apply_kernel(float* __restrict__ x, const float* __restrict__ stats,
                     const float* __restrict__ g, const float* __restrict__ b, int Nrows)
{
  const int i = blockIdx.x * blockDim.x + threadIdx.x;
  if (i >= Nrows * HF) return;
  const int c = i & (HF - 1);
  const float invN = 1.0f / (float)Nrows;
  const float mu  = stats[c] * invN;
  const float var = stats[HF + c] * invN - mu * mu;
  float y = (x[i] - mu) * rsqrtf(var + BN_EPS) * g[c] + b[c];
  x[i] = (y > 0.0f) ? y : 0.01f * y;              // LeakyReLU 0.01
}

// ------------------------- GAT attention scores ----------------------------
__global__ __launch_bounds__(256)
void gat_scores_kernel(const float* __restrict__ f, const float* __restrict__ al,
                       const float* __restrict__ ar, float* __restrict__ el,
                       float* __restrict__ er, int H)
{
  const int t = blockIdx.x * blockDim.x + threadIdx.x;
  if (t >= NNODES * H) return;
  const int n = t / H;
  const int h = t - n * H;
  const float* __restrict__ fr  = f + (size_t)n * H * HF + (size_t)h * HF;
  const float* __restrict__ alr = al + h * HF;
  const float* __restrict__ arr = ar + h * HF;
  float sl = 0.0f, sr = 0.0f;
#pragma unroll 8
  for (int d = 0; d < HF; ++d) {
    const float v = fr[d];
    sl += v * alr[d];
    sr += v * arr[d];
  }
  el[t] = sl;
  er[t] = sr;
}

// ------------------------- helpers -----------------------------------------
__global__ __launch_bounds__(256)
void init_u32_kernel(unsigned* __restrict__ p, unsigned val, int n)
{
  const int i = blockIdx.x * blockDim.x + threadIdx.x;
  if (i < n) p[i] = val;
}

__device__ __forceinline__ unsigned ord_encode(float x) {
  unsigned u = __float_as_uint(x);
  return (u & 0x80000000u) ? ~u : (u | 0x80000000u);
}
__device__ __forceinline__ float ord_decode(unsigned u) {
  return (u & 0x80000000u) ? __uint_as_float(u & 0x7fffffffu)
                           : __uint_as_float(~u);
}

// ------------------------- edge passes -------------------------------------
__global__ __launch_bounds__(256)
void edge_max_kernel(const float* __restrict__ el, const float* __restrict__ er,
                     const int* __restrict__ src, const int* __restrict__ dst,
                     float* __restrict__ elog, unsigned* __restrict__ menc,
                     int H, int logH)
{
  const int t = blockIdx.x * blockDim.x + threadIdx.x;
  if (t >= NEDGES * H) return;
  const int e = t >> logH;
  const int h = t & (H - 1);
  float x = el[src[e] * H + h] + er[dst[e] * H + h];
  x = (x > 0.0f) ? x : 0.2f * x;                  // attention LeakyReLU 0.2
  elog[t] = x;
  atomicMax(&menc[dst[e] * H + h], ord_encode(x));
}

__global__ __launch_bounds__(256)
void edge_exp_kernel(float* __restrict__ elog, const unsigned* __restrict__ menc,
                     const int* __restrict__ dst, float* __restrict__ denom,
                     int H, int logH)
{
  const int t = blockIdx.x * blockDim.x + threadIdx.x;
  if (t >= NEDGES * H) return;
  const int e = t >> logH;
  const int h = t & (H - 1);
  float mm = ord_decode(menc[dst[e] * H + h]);
  if (!isfinite(mm)) mm = 0.0f;
  const float ex = expf(elog[t] - mm);
  elog[t] = ex;
  atomicAdd(&denom[dst[e] * H + h], ex);
}

// out[dst] += f[src] * alpha.  One wave per edge, float4 gathers,
// per-head alpha computed in lanes 0..H-1 and shfl-broadcast.
__global__ __launch_bounds__(256)
void edge_scatter_kernel(const float* __restrict__ f, const float* __restrict__ elog,
                         const float* __restrict__ denom,
                         const int* __restrict__ src, const int* __restrict__ dst,
                         float* __restrict__ out, int H, int hd)
{
  const int w    = (blockIdx.x * blockDim.x + threadIdx.x) >> 5;  // edge id
  const int lane = threadIdx.x & 31;
  if (w >= NEDGES) return;
  const int s = src[w];
  const int d = dst[w];

  float a_h = 0.0f;
  if (lane < H)
    a_h = elog[(size_t)w * H + lane] / denom[(size_t)d * H + lane];

  const float4* __restrict__ frow = (const float4*)(f + (size_t)s * hd);
  float* __restrict__ orow = out + (size_t)d * hd;
  const int nq = hd >> 2;                          // float4s per row
  for (int j = lane; j < nq; j += 32) {
    const float4 fv = frow[j];
    const int h = j >> 4;                          // D = 64 -> 16 float4 / head
    const float a = __shfl(a_h, h, 32);
    atomicAdd(orow + 4 * j + 0, fv.x * a);
    atomicAdd(orow + 4 * j + 1, fv.y * a);
    atomicAdd(orow + 4 * j + 2, fv.z * a);
    atomicAdd(orow + 4 * j + 3, fv.w * a);
  }
}

__global__ __launch_bounds__(256)
void bias_leaky_kernel(float* __restrict__ x, const float* __restrict__ b,
                       int total, int mask)
{
  const int i = blockIdx.x * blockDim.x + threadIdx.x;
  if (i >= total) return;
  const float y = x[i] + b[i & mask];
  x[i] = (y > 0.0f) ? y : 0.01f * y;
}

// ------------------------- final classifier --------------------------------
__global__ __launch_bounds__(256)
void fc_kernel(const float* __restrict__ h, const float* __restrict__ Wf,
               const float* __restrict__ bf, float* __restrict__ out)
{
  const int n = blockIdx.x * blockDim.x + threadIdx.x;
  if (n >= NNODES) return;
  float a0 = bf[0], a1 = bf[1];
  const float* __restrict__ hr = h + (size_t)n * HF;
#pragma unroll 8
  for (int d = 0; d < HF; ++d) {
    const float v = hr[d];
    a0 += v * Wf[2 * d + 0];
    a1 += v * Wf[2 * d + 1];
  }
  out[2 * n + 0] = a0;
  out[2 * n + 1] = a1;
}

// ------------------------- host-side GAT layer -----------------------------
template<int KIN, int HD_, int H_>
static void run_gat(const float* cur, const float* W, const float* al,
                    const float* ar, const float* bias,
                    float* fbuf, float* obuf,
                    float* el, float* er, unsigned* menc, float* denom, float* elog,
                    const int* src, const int* dst, hipStream_t stream)
{
  // f = cur @ W
  gemm_wmma<KIN, HD_><<<dim3(NNODES / 16, HD_ / 64), 32, 0, stream>>>(
      cur, W, nullptr, fbuf);
  gat_scores_kernel<<<(NNODES * H_ + 255) / 256, 256, 0, stream>>>(
      fbuf, al, ar, el, er, H_);
  init_u32_kernel<<<(NNODES * H_ + 255) / 256, 256, 0, stream>>>(
      menc, 0x007FFFFFu /* ord_encode(-inf) */, NNODES * H_);
  hipMemsetAsync(denom, 0, (size_t)NNODES * H_ * sizeof(float), stream);
  hipMemsetAsync(obuf, 0, (size_t)NNODES * HD_ * sizeof(float), stream);
  const int logH = (H_ == 4) ? 2 : 0;
  edge_max_kernel<<<(NEDGES * H_ + 255) / 256, 256, 0, stream>>>(
      el, er, src, dst, elog, menc, H_, logH);
  edge_exp_kernel<<<(NEDGES * H_ + 255) / 256, 256, 0, stream>>>(
      elog, menc, dst, denom, H_, logH);
  edge_scatter_kernel<<<NEDGES / 8, 256, 0, stream>>>(
      fbuf, elog, denom, src, dst, obuf, H_, HD_);
  bias_leaky_kernel<<<(NNODES * HD_ + 255) / 256, 256, 0, stream>>>(
      obuf, bias, NNODES * HD_, HD_ - 1);
}

// ------------------------- entry point -------------------------------------
extern "C" void kernel_launch(void* const* d_in, const int* in_sizes, int n_in,
                              void* d_out, int out_size, void* d_ws, size_t ws_size,
                              hipStream_t stream)
{
  const float* feat    = (const float*)d_in[0];
  const int*   src     = (const int*)  d_in[1];
  const int*   dst     = (const int*)  d_in[2];
  const float* enc1_W  = (const float*)d_in[3];
  const float* enc1_b  = (const float*)d_in[4];
  const float* bn1_g   = (const float*)d_in[5];
  const float* bn1_b   = (const float*)d_in[6];
  const float* enc2_W  = (const float*)d_in[7];
  const float* enc2_b  = (const float*)d_in[8];
  const float* bn2_g   = (const float*)d_in[9];
  const float* bn2_b   = (const float*)d_in[10];
  const float* gat0_W  = (const float*)d_in[11];
  const float* gat0_al = (const float*)d_in[12];
  const float* gat0_ar = (const float*)d_in[13];
  const float* gat0_b  = (const float*)d_in[14];
  const float* gatm_W  = (const float*)d_in[15];
  const float* gatm_al = (const float*)d_in[16];
  const float* gatm_ar = (const float*)d_in[17];
  const float* gatm_b  = (const float*)d_in[18];
  const float* gat4_W  = (const float*)d_in[19];
  const float* gat4_al = (const float*)d_in[20];
  const float* gat4_ar = (const float*)d_in[21];
  const float* gat4_b  = (const float*)d_in[22];
  const float* fc_W    = (const float*)d_in[23];
  const float* fc_b    = (const float*)d_in[24];

  // workspace layout (floats)
  const size_t BUF = (size_t)NNODES * HDIM;            // 5.12 M floats
  float* ws    = (float*)d_ws;
  float* bufA  = ws;                                   // activations / out
  float* bufF  = ws + BUF;                             // projected features f
  float* bufB  = ws + 2 * BUF;                         // activations / out
  float*    el    = ws + 3 * BUF;
  float*    er    = el + (size_t)NNODES * NHEADS;
  unsigned* menc  = (unsigned*)(er + (size_t)NNODES * NHEADS);
  float*    denom = (float*)menc + (size_t)NNODES * NHEADS;
  float*    elog  = denom + (size_t)NNODES * NHEADS;
  float*    stats = elog + (size_t)NEDGES * NHEADS;    // 2*HF floats

  // ---- encoder 1: h = leaky(BN(feat @ W1 + b1)) -> bufA [N,64]
  gemm_wmma<FIN, HF><<<dim3(NNODES / 16, 1), 32, 0, stream>>>(
      feat, enc1_W, enc1_b, bufA);
  hipMemsetAsync(stats, 0, 2 * HF * sizeof(float), stream);
  bn_stats_kernel<<<256, HF, 0, stream>>>(bufA, stats, NNODES);
  bn_apply_kernel<<<(NNODES * HF + 255) / 256, 256, 0, stream>>>(
      bufA, stats, bn1_g, bn1_b, NNODES);

  // ---- encoder 2 -> bufB [N,64]
  gemm_wmma<HF, HF><<<dim3(NNODES / 16, 1), 32, 0, stream>>>(
      bufA, enc2_W, enc2_b, bufB);
  hipMemsetAsync(stats, 0, 2 * HF * sizeof(float), stream);
  bn_stats_kernel<<<256, HF, 0, stream>>>(bufB, stats, NNODES);
  bn_apply_kernel<<<(NNODES * HF + 255) / 256, 256, 0, stream>>>(
      bufB, stats, bn2_g, bn2_b, NNODES);

  float* cur = bufB;
  float* out = bufA;

  // ---- gat0: 64 -> 4x64
  run_gat<HF, HDIM, NHEADS>(cur, gat0_W, gat0_al, gat0_ar, gat0_b,
                            bufF, out, el, er, menc, denom, elog, src, dst, stream);
  { float* t = cur; cur = out; out = t; }

  // ---- 3x gatm: 256 -> 4x64
  for (int i = 0; i < 3; ++i) {
    run_gat<HDIM, HDIM, NHEADS>(cur, gatm_W + (size_t)i * HDIM * HDIM,
                                gatm_al + (size_t)i * NHEADS * HF,
                                gatm_ar + (size_t)i * NHEADS * HF,
                                gatm_b + (size_t)i * HDIM,
                                bufF, out, el, er, menc, denom, elog, src, dst, stream);
    { float* t = cur; cur = out; out = t; }
  }

  // ---- gat4: 256 -> 1x64
  run_gat<HDIM, HF, 1>(cur, gat4_W, gat4_al, gat4_ar, gat4_b,
                       bufF, out, el, er, menc, denom, elog, src, dst, stream);
  { float* t = cur; cur = out; out = t; }

  // ---- classifier: [N,64] @ [64,2] + b -> d_out
  fc_kernel<<<(NNODES + 255) / 256, 256, 0, stream>>>(
      cur, fc_W, fc_b, (float*)d_out);
}